// KmeansEuclideanObjectTokens_6768868458546
// MI455X (gfx1250) — compile-verified
//
#include <hip/hip_runtime.h>

// Problem constants (from reference setup_inputs)
#define B_   16
#define N_   9216      // 96*96
#define C_   256
#define K_   21
#define KP   32        // K padded to 2 WMMA column tiles
#define ITERS 20
#define BLOCKS_PER_BATCH 16
#define ROWS_PER_BLOCK  (N_/BLOCKS_PER_BATCH)   // 576
#define TILES_PER_BLOCK (ROWS_PER_BLOCK/16)     // 36
#define NWAVES 8

typedef __attribute__((ext_vector_type(2))) float v2f;
typedef __attribute__((ext_vector_type(8))) float v8f;
typedef __attribute__((ext_vector_type(4))) unsigned int v4u;
typedef __attribute__((ext_vector_type(4))) int v4i;
typedef __attribute__((ext_vector_type(8))) int v8i;

__global__ void zero_ws_kernel(float* __restrict__ p, int n) {
    int i = blockIdx.x * blockDim.x + threadIdx.x;
    if (i < n) p[i] = 0.f;
}

// cent0[b][k][:] = x[b][init_idx[b][k]][:]
__global__ void gather_init_kernel(const float* __restrict__ x,
                                   const int* __restrict__ idx,
                                   float* __restrict__ cent) {
    const int k = blockIdx.x, b = blockIdx.y, c = threadIdx.x;
    const int row = idx[b * K_ + k];
    cent[((size_t)(b * K_ + k)) * C_ + c] = x[((size_t)b * N_ + row) * C_ + c];
}

// monotone float -> uint mapping (preserves < ordering)
__device__ __forceinline__ unsigned sortable_f32(float f) {
    unsigned u = __float_as_uint(f);
    return (u & 0x80000000u) ? ~u : (u | 0x80000000u);
}

// Phase 1: assignment via WMMA fp32 distance matmul + hierarchical scatter-add
__global__ void __launch_bounds__(256)
assign_accum_kernel(const float* __restrict__ x, const float* __restrict__ cent,
                    float* __restrict__ sums, float* __restrict__ cnts) {
    // centroids transposed, channel-pair interleaved: idx(c,k) = ((c>>1)*KP + k)*2 + (c&1)
    __shared__ float sCT[(C_ / 2) * KP * 2];   // 32 KB  (WMMA B operand layout)
    __shared__ float sSq[KP];
    __shared__ float sSum[K_ * C_];            // 21.5 KB: TDM staging, then local scatter-add
    __shared__ float sCnt[K_];
    __shared__ int   sAsn[NWAVES][16];

    const int tid  = threadIdx.x;
    const int b    = blockIdx.y;
    const int wv   = tid >> 5;
    const int lane = tid & 31;

    const float* cb = cent + (size_t)b * K_ * C_;

    // ---- stage this batch's centroid block into LDS via the Tensor Data Mover ----
    // D# per CDNA5 ISA: 2D tensor, data_size=4B,
    // tensor_dim0=C_, tensor_dim1=K_, tile=C_ x K_, stride0=C_, type=2.
    if (wv == 0) {
        unsigned lds0 = (unsigned)(size_t)(void*)sSum;   // low 32 bits of flat pointer = LDS byte offset
        unsigned long long ga = (unsigned long long)(size_t)(const void*)cb;
        v4u g0 = { 1u,                                   // count=1 (valid descriptor)
                   lds0,                                 // lds_addr
                   (unsigned)ga,                         // global_addr[31:0]
                   (unsigned)(ga >> 32) | (2u << 30) };  // global_addr[56:32] | type=2
        v8i g1 = { 0x2 << 16,                            // data_size=2 (4 bytes)
                   (C_ & 0xFFFF) << 16,                  // tensor_dim0[15:0]
                   (C_ >> 16) | ((K_ & 0xFFFF) << 16),   // tensor_dim0[31:16] | tensor_dim1[15:0]
                   (K_ >> 16) | (C_ << 16),              // tensor_dim1[31:16] | tile_dim0
                   K_,                                   // tile_dim1 (tile_dim2=0)
                   C_,                                   // tensor_dim0_stride[31:0]
                   0, 0 };
        v4i gz = { 0, 0, 0, 0 };
#if __clang_major__ >= 23
        v8i gz8 = { 0, 0, 0, 0, 0, 0, 0, 0 };
        __builtin_amdgcn_tensor_load_to_lds(g0, g1, gz, gz, gz8, 0);
#else
        __builtin_amdgcn_tensor_load_to_lds(g0, g1, gz, gz, 0);
#endif
        __builtin_amdgcn_s_wait_tensorcnt(0);
    }
    __syncthreads();

    // ---- transpose staged centroids (row-major [k][c]) into WMMA B layout + pad ----
    for (int i = tid; i < K_ * C_; i += 256) {
        int k = i / C_, c = i % C_;
        sCT[((c >> 1) * KP + k) * 2 + (c & 1)] = sSum[i];
    }
    for (int i = tid; i < (KP - K_) * C_; i += 256) {   // pad columns k=21..31 -> 0
        int k = K_ + i / C_, c = i % C_;
        sCT[((c >> 1) * KP + k) * 2 + (c & 1)] = 0.f;
    }
    __syncthreads();

    // ---- zero local accumulators; centroid squared norms ----
    for (int i = tid; i < K_ * C_; i += 256) sSum[i] = 0.f;
    if (tid < K_) sCnt[tid] = 0.f;
    if (tid < KP) {        // padded columns get +huge so argmin ignores them
        float s = 3.0e38f;
        if (tid < K_) {
            s = 0.f;
            for (int c = 0; c < C_; ++c) {
                float v = sCT[((c >> 1) * KP + tid) * 2 + (c & 1)];
                s += v * v;
            }
        }
        sSq[tid] = s;
    }
    __syncthreads();

    const int col = lane & 15;     // WMMA C/D column (= centroid index within tile)
    const int kp2 = lane >> 4;     // A/B fragment: channel-pair select per ISA layout
    const float cSq0 = sSq[col];
    const float cSq1 = sSq[16 + col];

    const float* xb = x + (size_t)b * N_ * C_;

    for (int tt = wv; tt < TILES_PER_BLOCK; tt += NWAVES) {
        const int n0 = blockIdx.x * ROWS_PER_BLOCK + tt * 16;
        const float* xrow = xb + (size_t)(n0 + col) * C_ + 2 * kp2;
        if (tt + NWAVES < TILES_PER_BLOCK)
            __builtin_prefetch(xrow + (size_t)NWAVES * 16 * C_, 0, 1);

        v8f acc0 = {};   // dot(x_row, c_k), k = col       (columns 0..15)
        v8f acc1 = {};   // dot(x_row, c_k), k = 16 + col  (columns 16..31)
        #pragma unroll 4
        for (int c0 = 0; c0 < C_; c0 += 4) {
            v2f a = *(const v2f*)(xrow + c0);                       // A: 16x4 fp32
            const int cidx = ((c0 >> 1) + kp2) * KP;
            v2f b0 = *(const v2f*)&sCT[(cidx + col) * 2];           // B: 4x16 fp32
            v2f b1 = *(const v2f*)&sCT[(cidx + 16 + col) * 2];
            acc0 = __builtin_amdgcn_wmma_f32_16x16x4_f32(false, a, false, b0,
                                                         (short)0, acc0, false, false);
            acc1 = __builtin_amdgcn_wmma_f32_16x16x4_f32(false, a, false, b1,
                                                         (short)0, acc1, false, false);
        }

        // ---- per-row argmin_k of (||c||^2 - 2 x.c) ; branchless u64-key reduction ----
        #pragma unroll
        for (int r = 0; r < 8; ++r) {   // lanes 0-15: row r ; lanes 16-31: row r+8
            float d0 = __builtin_fmaf(-2.f, acc0[r], cSq0);
            float d1 = __builtin_fmaf(-2.f, acc1[r], cSq1);
            // key = sortable(dist):k  -> min gives argmin with lowest-k tie-break
            unsigned long long k0 = ((unsigned long long)sortable_f32(d0) << 32) | (unsigned)col;
            unsigned long long k1 = ((unsigned long long)sortable_f32(d1) << 32) | (unsigned)(16 + col);
            unsigned long long key = k1 < k0 ? k1 : k0;
            #pragma unroll
            for (int off = 1; off < 16; off <<= 1) {    // reduce within 16-lane halves
                unsigned long long o = __shfl_xor(key, off, 32);
                key = o < key ? o : key;
            }
            int bk = (int)(key & 0x3F);
            if (lane == 0)  sAsn[wv][r]     = bk;
            if (lane == 16) sAsn[wv][8 + r] = bk;
        }
        __builtin_amdgcn_wave_barrier();   // same-wave LDS ops are in-order

        // ---- scatter-add rows into LDS partials ----
        if (lane < 16) atomicAdd(&sCnt[sAsn[wv][lane]], 1.f);
        #pragma unroll 1
        for (int r = 0; r < 16; ++r) {
            const int ka = sAsn[wv][r];
            const float* xr = xb + (size_t)(n0 + r) * C_;
            #pragma unroll
            for (int c = lane; c < C_; c += 32)
                atomicAdd(&sSum[ka * C_ + c], xr[c]);
        }
    }
    __syncthreads();

    // ---- single global flush per block (tiny atomic traffic) ----
    float* gs = sums + (size_t)b * K_ * C_;
    for (int i = tid; i < K_ * C_; i += 256) atomicAdd(&gs[i], sSum[i]);
    if (tid < K_) atomicAdd(&cnts[b * K_ + tid], sCnt[tid]);
}

// Phase 2: cent = counts>0 ? sums/max(counts,1) : cent ; reset accumulators
__global__ void update_cent_kernel(float* __restrict__ cent,
                                   float* __restrict__ sums,
                                   float* __restrict__ cnts) {
    const int k = blockIdx.x, b = blockIdx.y, c = threadIdx.x;
    const size_t i = ((size_t)(b * K_ + k)) * C_ + c;
    const float cnt = cnts[b * K_ + k];
    const float s = sums[i];
    const float old = cent[i];
    cent[i] = (cnt > 0.f) ? (s / fmaxf(cnt, 1.f)) : old;
    sums[i] = 0.f;
    if (c == 0) cnts[b * K_ + k] = 0.f;
}

extern "C" void kernel_launch(void* const* d_in, const int* in_sizes, int n_in,
                              void* d_out, int out_size, void* d_ws, size_t ws_size,
                              hipStream_t stream) {
    const float* x   = (const float*)d_in[0];
    const int*   idx = (const int*)d_in[1];

    float* cent = (float*)d_ws;                       // [B,K,C]
    float* sums = cent + (size_t)B_ * K_ * C_;        // [B,K,C]
    float* cnts = sums + (size_t)B_ * K_ * C_;        // [B,K]

    const int nzero = B_ * K_ * C_ + B_ * K_;         // sums + cnts contiguous
    zero_ws_kernel<<<(nzero + 255) / 256, 256, 0, stream>>>(sums, nzero);
    gather_init_kernel<<<dim3(K_, B_), C_, 0, stream>>>(x, idx, cent);

    for (int it = 0; it < ITERS; ++it) {
        assign_accum_kernel<<<dim3(BLOCKS_PER_BATCH, B_), 256, 0, stream>>>(x, cent, sums, cnts);
        update_cent_kernel<<<dim3(K_, B_), C_, 0, stream>>>(cent, sums, cnts);
    }

    hipMemcpyAsync(d_out, cent, (size_t)B_ * K_ * C_ * sizeof(float),
                   hipMemcpyDeviceToDevice, stream);
}